// LBFGS_59304908423828
// MI455X (gfx1250) — compile-verified
//
#include <hip/hip_runtime.h>

typedef float v2f __attribute__((ext_vector_type(2)));
typedef float v8f __attribute__((ext_vector_type(8)));

constexpr int Mh = 32;    // history length
constexpr int Bn = 64;    // batch
constexpr int Dn = 8192;  // feature dim
constexpr int CROWS = 64; // gram rows: [s_0..s_31, y_0..y_31]
constexpr int CCOLS = 48; // gram cols: [y_0..y_31, q0=-frc, 15 zero cols]
constexpr size_t C_ELEMS = (size_t)Bn * CROWS * CCOLS; // 196608 floats
constexpr size_t W_STRIDE = 72;                        // per-batch weights: wy[32], wz[32], g

// ---------------------------------------------------------------------------
// Kernel A: per-batch Gram block via V_WMMA_F32_16X16X4_F32.
//   C[i][j]      = A_row_i . B_col_j  over K = 8192
//   A rows 0..31 = s_i[b], rows 32..63 = y_i[b]
//   B cols 0..31 = y_j[b], col 32 = q0 = -frc[b], cols 33..47 = 0
// One workgroup (12 waves) per batch; one 16x16 tile per wave (4x3 tile grid).
// The tile-column branch is wave-uniform and fully hoisted: each K-loop body
// is branch-free so no exec-mask churn surrounds the WMMAs.
// ---------------------------------------------------------------------------
__global__ __launch_bounds__(384)
void lbfgs_gram_wmma(const float* __restrict__ s, const float* __restrict__ y,
                     const float* __restrict__ frc, float* __restrict__ cfull) {
#if defined(__gfx1250__)
  const int b    = blockIdx.x;
  const int wave = threadIdx.x >> 5;
  const int lane = threadIdx.x & 31;
  const int half = lane >> 4;   // lane group: selects K sub-pair
  const int lr   = lane & 15;   // row/col within tile
  const int ti   = wave / 3;    // 0..3 (row tile)
  const int tj   = wave % 3;    // 0..2 (col tile)
  const int ko   = half * 2;    // K offset for this lane group

  // A-matrix row for this lane (ISA 16x4 f32 A layout: lanes 0-15 -> K=0,1;
  // lanes 16-31 -> K=2,3; both VGPRs hold M=lane%16).
  const int gi = ti * 16 + lr;
  const float* arow = (gi < Mh)
      ? (s + ((size_t)gi * Bn + b) * Dn)
      : (y + ((size_t)(gi - Mh) * Bn + b) * Dn);

  v8f c = {0.f, 0.f, 0.f, 0.f, 0.f, 0.f, 0.f, 0.f};

  if (tj < 2) {
    // Pure GEMM tiles: B[k][n] = Y_col(tj*16+n)[k]; B-row striping matches
    // C/D (VGPR0 = K rows 0/2, VGPR1 = K rows 1/3) -> same float2 pattern as A.
    const float* ap = arow + ko;
    const float* bp = y + ((size_t)(tj * 16 + lr) * Bn + b) * Dn + ko;
    #pragma unroll 8
    for (int k = 0; k < Dn; k += 4) {
      const v2f a2 = *(const v2f*)ap;
      const v2f b2 = *(const v2f*)bp;
      ap += 4;
      bp += 4;
      c = __builtin_amdgcn_wmma_f32_16x16x4_f32(false, a2, false, b2,
                                                (short)0, c, false, false);
    }
  } else {
    // q0 column tile: only tile-col 0 is live (q0 = -frc), cols 1..15 are 0.
    // All lanes issue the (lane-invariant) frc load; the loop-invariant
    // lr==0 predicate selects via cndmask, keeping the loop branch-free.
    const bool live = (lr == 0);
    const float* ap = arow + ko;
    const float* fp = frc + (size_t)b * Dn + ko;
    #pragma unroll 8
    for (int k = 0; k < Dn; k += 4) {
      const v2f a2 = *(const v2f*)ap;
      const v2f fv = *(const v2f*)fp;
      ap += 4;
      fp += 4;
      v2f b2;
      b2.x = live ? -fv.x : 0.f;
      b2.y = live ? -fv.y : 0.f;
      c = __builtin_amdgcn_wmma_f32_16x16x4_f32(false, a2, false, b2,
                                                (short)0, c, false, false);
    }
  }

  // C/D layout: VGPR r -> row M = r (lanes 0-15) / M = r+8 (lanes 16-31), col = lane%16.
  float* cb = cfull + (size_t)b * (CROWS * CCOLS);
  #pragma unroll
  for (int r = 0; r < 8; ++r) {
    const int row = ti * 16 + half * 8 + r;
    const int col = tj * 16 + lr;
    cb[row * CCOLS + col] = c[r];
  }
#endif
}

// ---------------------------------------------------------------------------
// Kernel B: per-batch triangular recurrences on the 32x32 Gram scalars.
//   P[i][j] = s_i.y_j ; G[i][j] = y_i.y_j ; u[i] = s_i.q0 ; v[i] = y_i.q0
//   backward: a_i = r_i (u_i - sum_{j>i} P_ij a_j)
//   g = P[31][31]/G[31][31]
//   forward:  b_i = r_i ( g(v_i - sum_j a_j G_ij) + sum_{j<i} (a_j-b_j) P_ji )
//   out weights: wy_i = g a_i ; wz_i = b_i - a_i ; plus g itself.
// ---------------------------------------------------------------------------
__global__ __launch_bounds__(32)
void lbfgs_solve(const float* __restrict__ cfull, float* __restrict__ wts) {
  __shared__ float P[32][32];
  __shared__ float G[32][32];
  __shared__ float u[32];
  __shared__ float v[32];
  const int b = blockIdx.x;
  const int t = threadIdx.x;
  const float* cb = cfull + (size_t)b * (CROWS * CCOLS);
  for (int j = 0; j < 32; ++j) {
    P[t][j] = cb[t * CCOLS + j];
    G[t][j] = cb[(32 + t) * CCOLS + j];
  }
  u[t] = cb[t * CCOLS + 32];
  v[t] = cb[(32 + t) * CCOLS + 32];
  __syncthreads();

  if (t == 0) {
    float r[32], a[32], cc[32];
    for (int i = 0; i < 32; ++i) r[i] = 1.0f / P[i][i];
    for (int i = 31; i >= 0; --i) {
      float acc = u[i];
      for (int j = i + 1; j < 32; ++j) acc -= P[i][j] * a[j];
      a[i] = r[i] * acc;
    }
    const float g = P[31][31] / G[31][31];
    float* w = wts + (size_t)b * W_STRIDE;
    for (int i = 0; i < 32; ++i) {
      float acc = v[i];
      for (int j = 0; j < 32; ++j) acc -= a[j] * G[i][j];
      acc *= g;
      for (int j = 0; j < i; ++j) acc += cc[j] * P[j][i]; // y_i.s_j = P[j][i]
      const float bi = r[i] * acc;
      cc[i]     = a[i] - bi;
      w[i]      = g * a[i];   // weight on y_i
      w[32 + i] = bi - a[i];  // weight on s_i
    }
    w[64] = g;                // weight on frc
  }
}

// ---------------------------------------------------------------------------
// Kernel C: out[b,:] = g*frc[b,:] + sum_i wy_i*y[i,b,:] + sum_i wz_i*s[i,b,:]
// float4-vectorized streaming combine; s+y (128 MB) fit in the 192 MB L2 so
// this second pass mostly avoids HBM.
// ---------------------------------------------------------------------------
__global__ __launch_bounds__(256)
void lbfgs_combine(const float* __restrict__ s, const float* __restrict__ y,
                   const float* __restrict__ frc, const float* __restrict__ wts,
                   float* __restrict__ out) {
  const int b     = blockIdx.x >> 3;
  const int chunk = blockIdx.x & 7;
  const int d4    = chunk * 256 + threadIdx.x;     // float4 index, 0..2047
  const float* w  = wts + (size_t)b * W_STRIDE;

  const float g = w[64];
  const float4 q = ((const float4*)(frc + (size_t)b * Dn))[d4];
  float4 acc = {g * q.x, g * q.y, g * q.z, g * q.w};

  #pragma unroll 4
  for (int i = 0; i < 32; ++i) {
    const float4 yv = ((const float4*)(y + ((size_t)i * Bn + b) * Dn))[d4];
    const float4 sv = ((const float4*)(s + ((size_t)i * Bn + b) * Dn))[d4];
    const float wy = w[i], wz = w[32 + i];
    acc.x += wy * yv.x + wz * sv.x;
    acc.y += wy * yv.y + wz * sv.y;
    acc.z += wy * yv.z + wz * sv.z;
    acc.w += wy * yv.w + wz * sv.w;
  }
  ((float4*)(out + (size_t)b * Dn))[d4] = acc;
}

// ---------------------------------------------------------------------------
extern "C" void kernel_launch(void* const* d_in, const int* in_sizes, int n_in,
                              void* d_out, int out_size, void* d_ws, size_t ws_size,
                              hipStream_t stream) {
  (void)in_sizes; (void)n_in; (void)out_size; (void)ws_size;
  const float* s   = (const float*)d_in[0];
  const float* y   = (const float*)d_in[1];
  const float* frc = (const float*)d_in[2];
  float* out   = (float*)d_out;
  float* cfull = (float*)d_ws;             // 64 * 64*48 floats
  float* wts   = cfull + C_ELEMS;          // 64 * 72 floats

  lbfgs_gram_wmma<<<Bn, 384, 0, stream>>>(s, y, frc, cfull);
  lbfgs_solve<<<Bn, 32, 0, stream>>>(cfull, wts);
  lbfgs_combine<<<Bn * 8, 256, 0, stream>>>(s, y, frc, wts, out);
}